// GAT_7842610282680
// MI455X (gfx1250) — compile-verified
//
#include <hip/hip_runtime.h>

// ---------------------------------------------------------------------------
// GAT (2-layer) for MI455X / gfx1250, wave32.
// GEMMs use V_WMMA_F32_16X16X4_F32 (fp32 in / fp32 acc): workload is
// L2-bandwidth bound (everything < 192MB L2), so full precision costs nothing.
// B is pre-packed into WMMA fragment order so the GEMM inner loop is
//   ds_load_b64 (A frag) + global_load_b64 (B frag) + v_wmma, no predication,
//   no 64-bit address math.
// ---------------------------------------------------------------------------

typedef __attribute__((ext_vector_type(2))) float v2f;
typedef __attribute__((ext_vector_type(8))) float v8f;

#define NEG_SLOPE 0.2f

// ---------------- utility: fill a u32 buffer (zeros floats / max-keys) -----
__global__ void fill_u32_kernel(unsigned* __restrict__ p, unsigned v, size_t n) {
  size_t i = (size_t)blockIdx.x * blockDim.x + threadIdx.x;
  if (i < n) p[i] = v;
}

// ---------------- pack B[K,N] into WMMA fragment order ---------------------
// Bp[nt][ks][lane][0..1]:
//   col = nt*16 + (lane&15), k = ks*4 + 2*(lane>>4)
//   Bp = { B[k][col], B[k+1][col] }   (zero when col >= N)
// In the GEMM each wave then reads one contiguous v2f per lane per k-step.
__global__ void pack_b_kernel(const float* __restrict__ B, float* __restrict__ Bp,
                              int N, int K) {
  const int ntiles = (N + 15) >> 4;
  const int ksteps = K >> 2;
  int idx = blockIdx.x * blockDim.x + threadIdx.x;   // (nt, ks, lane)
  if (idx >= ntiles * ksteps * 32) return;
  int lane = idx & 31;
  int ks   = (idx >> 5) % ksteps;
  int nt   = (idx >> 5) / ksteps;
  int col  = nt * 16 + (lane & 15);
  int k    = ks * 4 + 2 * (lane >> 4);
  float v0 = 0.f, v1 = 0.f;
  if (col < N) {
    v0 = B[k * N + col];
    v1 = B[(k + 1) * N + col];
  }
  Bp[idx * 2 + 0] = v0;
  Bp[idx * 2 + 1] = v1;
}

// ---------------- GEMM: C[M,N] = A[M,K] * Bp(packed) -----------------------
// M % 16 == 0, K % 4 == 0. 128 threads = 4 waves per block.
// Block computes a 16-row stripe; waves round-robin over 16-col tiles.
// A tile staged in LDS with padded stride -> conflict-free ds_load_b64.
__global__ __launch_bounds__(128) void gemm_f32_wmma(
    const float* __restrict__ A, const float* __restrict__ Bp,
    float* __restrict__ C, int N, int K) {
  __shared__ float As[16 * (256 + 4)];      // supports K <= 256
  const int LDA  = K + 4;                   // pad: no LDS bank conflicts
  const int row0 = blockIdx.x * 16;
  const int tid  = threadIdx.x;

  // cooperative, coalesced float4 copy of the 16 x K A tile into LDS
  const int q = K >> 2;                     // float4s per row
  for (int idx = tid; idx < 16 * q; idx += 128) {
    int r = idx / q, c = idx % q;
    float4 val = ((const float4*)(A + (size_t)(row0 + r) * K))[c];
    *(float4*)&As[r * LDA + c * 4] = val;
  }
  __syncthreads();

  const int wave = tid >> 5;
  const int lane = tid & 31;
  const int lm   = lane & 15;               // row (A) / col (B,C) within tile
  const int hi   = lane >> 4;               // half-wave selector
  const int koff = 2 * hi;                  // A frag: VGPR0 K=0|2, VGPR1 K=1|3
  const int ntiles = (N + 15) >> 4;
  const int ksteps = K >> 2;

  for (int nt = wave; nt < ntiles; nt += 4) {   // uniform per wave: EXEC all-1s
    const float* bp = Bp + ((nt * ksteps) * 32 + lane * 2);
    int aoff = lm * LDA + koff;
    v8f acc = {};
    #pragma unroll 4
    for (int ks = 0; ks < ksteps; ++ks) {
      v2f a = *(const v2f*)&As[aoff];          // ds_load_b64
      v2f b = *(const v2f*)bp;                 // global_load_b64, coalesced
      acc = __builtin_amdgcn_wmma_f32_16x16x4_f32(false, a, false, b,
                                                  (short)0, acc, false, false);
      aoff += 4;
      bp   += 64;
    }
    const int col = nt * 16 + lm;
    if (col < N) {
      // D layout: VGPR v -> row (v + 8*hi), col = lane%16
      unsigned cbase = (unsigned)(row0 + hi * 8) * (unsigned)N + (unsigned)col;
      #pragma unroll
      for (int v = 0; v < 8; ++v) C[cbase + (unsigned)(v * N)] = acc[v];
    }
  }
}

// ---------------- attention coefficients (wave per node, coalesced) --------
// el/er[n,h] = <z[n,h,:], a[h,:]>; al/ar flat over H*F matches z row layout.
__global__ __launch_bounds__(256) void attn_coef_kernel(
    const float* __restrict__ z, const float* __restrict__ al,
    const float* __restrict__ ar, float* __restrict__ el, float* __restrict__ er,
    int n_nodes, int H, int F) {
  int n    = (blockIdx.x * blockDim.x + threadIdx.x) >> 5;
  int lane = threadIdx.x & 31;
  if (n >= n_nodes) return;
  const int HF = H * F;
  const float* zp = z + (size_t)n * HF;
  float sl0 = 0.f, sr0 = 0.f, sl1 = 0.f, sr1 = 0.f;
  for (int f = lane; f < HF; f += 32) {
    float v = zp[f];
    float p = v * al[f];
    float q = v * ar[f];
    if (f < F) { sl0 += p; sr0 += q; } else { sl1 += p; sr1 += q; }
  }
  #pragma unroll
  for (int off = 16; off; off >>= 1) {
    sl0 += __shfl_xor(sl0, off, 32);
    sr0 += __shfl_xor(sr0, off, 32);
    sl1 += __shfl_xor(sl1, off, 32);
    sr1 += __shfl_xor(sr1, off, 32);
  }
  if (lane == 0) {
    el[n * H] = sl0;
    er[n * H] = sr0;
    if (H > 1) { el[n * H + 1] = sl1; er[n * H + 1] = sr1; }
  }
}

// ---------------- monotone float<->uint map for atomic float max -----------
__device__ __forceinline__ unsigned f2ord(float f) {
  unsigned u = __float_as_uint(f);
  return (u & 0x80000000u) ? ~u : (u | 0x80000000u);
}
__device__ __forceinline__ float ord2f(unsigned u) {
  return __uint_as_float((u & 0x80000000u) ? (u ^ 0x80000000u) : ~u);
}

// ---------------- edge pass 1: score + segment max via atomic_max_u32 ------
__global__ void edge_score_kernel(const int* __restrict__ src,
                                  const int* __restrict__ dst,
                                  const float* __restrict__ el,
                                  const float* __restrict__ er,
                                  float* __restrict__ ev, unsigned* __restrict__ mk,
                                  int n_edges, int H) {
  int idx = blockIdx.x * blockDim.x + threadIdx.x;   // (edge, head)
  if (idx >= n_edges * H) return;
  int e = idx / H, h = idx % H;
  int s = src[e], d = dst[e];
  float v = el[s * H + h] + er[d * H + h];
  v = v > 0.f ? v : NEG_SLOPE * v;                   // leaky relu
  ev[idx] = v;
  atomicMax(&mk[d * H + h], f2ord(v));               // global_atomic_max_u32
}

// ---------------- edge pass 2: exp(e - m[dst]) + segment sum ---------------
__global__ void edge_exp_kernel(const int* __restrict__ dst,
                                const unsigned* __restrict__ mk,
                                float* __restrict__ ev, float* __restrict__ denom,
                                int n_edges, int H) {
  int idx = blockIdx.x * blockDim.x + threadIdx.x;
  if (idx >= n_edges * H) return;
  int e = idx / H, h = idx % H;
  int d = dst[e];
  float m = ord2f(mk[d * H + h]);
  float a = __expf(ev[idx] - m);
  ev[idx] = a;
  atomicAdd(&denom[d * H + h], a);                   // global_atomic_add_f32
}

// ---------------- edge pass 3: agg[dst] += alpha * z[src] ------------------
// one wave per edge; lanes stride the H*F contiguous feature row (coalesced).
__global__ __launch_bounds__(256) void edge_aggregate_kernel(
    const int* __restrict__ src, const int* __restrict__ dst,
    const float* __restrict__ z, const float* __restrict__ ev,
    const float* __restrict__ denom, float* __restrict__ agg,
    int n_edges, int H, int F) {
  int gw   = (blockIdx.x * blockDim.x + threadIdx.x) >> 5;  // wave32
  int lane = threadIdx.x & 31;
  if (gw >= n_edges) return;
  int s = src[gw], d = dst[gw];
  float a0 = ev[gw * H] / denom[d * H];
  float a1 = (H > 1) ? ev[gw * H + 1] / denom[d * H + 1] : 0.f;
  const int HF = H * F;
  const float* zs = z + (size_t)s * HF;              // L2-resident gather
  float* ag = agg + (size_t)d * HF;
  for (int f = lane; f < HF; f += 32) {
    float alpha = (f < F) ? a0 : a1;
    atomicAdd(&ag[f], zs[f] * alpha);
  }
}

// ---------------- in-place ELU(x + b) --------------------------------------
__global__ void elu_bias_kernel(float* __restrict__ x, const float* __restrict__ b,
                                size_t n, int HF) {
  size_t i = (size_t)blockIdx.x * blockDim.x + threadIdx.x;
  if (i >= n) return;
  float v = x[i] + b[i % HF];
  x[i] = v > 0.f ? v : (__expf(v) - 1.f);
}

// ---------------- out += bias ----------------------------------------------
__global__ void bias_add_kernel(float* __restrict__ x, const float* __restrict__ b,
                                size_t n, int Cc) {
  size_t i = (size_t)blockIdx.x * blockDim.x + threadIdx.x;
  if (i >= n) return;
  x[i] += b[i % Cc];
}

// ===========================================================================
extern "C" void kernel_launch(void* const* d_in, const int* in_sizes, int n_in,
                              void* d_out, int out_size, void* d_ws, size_t ws_size,
                              hipStream_t stream) {
  const float* X   = (const float*)d_in[0];   // [N, 256]
  const float* W1  = (const float*)d_in[1];   // [256, 200]
  const float* al1 = (const float*)d_in[2];   // [2, 100]
  const float* ar1 = (const float*)d_in[3];   // [2, 100]
  const float* b1  = (const float*)d_in[4];   // [200]
  const float* W2  = (const float*)d_in[5];   // [200, 32]
  const float* al2 = (const float*)d_in[6];   // [1, 32]
  const float* ar2 = (const float*)d_in[7];   // [1, 32]
  const float* b2  = (const float*)d_in[8];   // [32]
  const int* src   = (const int*)d_in[9];     // [E]
  const int* dst   = (const int*)d_in[10];    // [E]
  float* out       = (float*)d_out;           // [N, 32]

  const int FIN = 256, H1 = 2, F1 = 100, HF1 = H1 * F1, C = 32;
  const int N = in_sizes[0] / FIN;            // 50000
  const int E = in_sizes[9];                  // 1600000

  const int nt1 = (HF1 + 15) / 16, ks1 = FIN / 4;   // 13 tiles, 64 k-steps
  const int nt2 = (C + 15) / 16,   ks2 = HF1 / 4;   // 2 tiles, 50 k-steps

  // ---- workspace layout (~108 MB; L2-friendly) ----
  char* w = (char*)d_ws;
  float*    z1   = (float*)w;    w += (size_t)N * HF1 * 4;   // 40 MB
  float*    agg1 = (float*)w;    w += (size_t)N * HF1 * 4;   // 40 MB (becomes h)
  float*    z2   = (float*)w;    w += (size_t)N * C * 4;     // 6.4 MB
  float*    ev1  = (float*)w;    w += (size_t)E * H1 * 4;    // 12.8 MB
  float*    ev2  = (float*)w;    w += (size_t)E * 4;         // 6.4 MB
  float*    Bp1  = (float*)w;    w += (size_t)nt1 * ks1 * 64 * 4;  // 208 KB
  float*    Bp2  = (float*)w;    w += (size_t)nt2 * ks2 * 64 * 4;  // 25 KB
  float*    el1  = (float*)w;    w += (size_t)N * H1 * 4;
  float*    er1  = (float*)w;    w += (size_t)N * H1 * 4;
  unsigned* mk1  = (unsigned*)w; w += (size_t)N * H1 * 4;
  float*    dn1  = (float*)w;    w += (size_t)N * H1 * 4;
  float*    el2  = (float*)w;    w += (size_t)N * 4;
  float*    er2  = (float*)w;    w += (size_t)N * 4;
  unsigned* mk2  = (unsigned*)w; w += (size_t)N * 4;
  float*    dn2  = (float*)w;    w += (size_t)N * 4;

  auto blk = [](size_t n, int t) { return (unsigned)((n + t - 1) / t); };

  // ---- init accumulators (key 0 == identity for ordered-float max) ----
  fill_u32_kernel<<<blk((size_t)N * HF1, 256), 256, 0, stream>>>((unsigned*)agg1, 0u, (size_t)N * HF1);
  fill_u32_kernel<<<blk((size_t)N * H1, 256), 256, 0, stream>>>(mk1, 0u, (size_t)N * H1);
  fill_u32_kernel<<<blk((size_t)N * H1, 256), 256, 0, stream>>>((unsigned*)dn1, 0u, (size_t)N * H1);
  fill_u32_kernel<<<blk((size_t)N, 256), 256, 0, stream>>>(mk2, 0u, (size_t)N);
  fill_u32_kernel<<<blk((size_t)N, 256), 256, 0, stream>>>((unsigned*)dn2, 0u, (size_t)N);
  fill_u32_kernel<<<blk((size_t)N * C, 256), 256, 0, stream>>>((unsigned*)out, 0u, (size_t)N * C);

  // ---- layer 1 ----
  pack_b_kernel<<<blk((size_t)nt1 * ks1 * 32, 256), 256, 0, stream>>>(W1, Bp1, HF1, FIN);
  gemm_f32_wmma<<<N / 16, 128, 0, stream>>>(X, Bp1, z1, HF1, FIN);
  attn_coef_kernel<<<blk((size_t)N * 32, 256), 256, 0, stream>>>(z1, al1, ar1, el1, er1, N, H1, F1);
  edge_score_kernel<<<blk((size_t)E * H1, 256), 256, 0, stream>>>(src, dst, el1, er1, ev1, mk1, E, H1);
  edge_exp_kernel<<<blk((size_t)E * H1, 256), 256, 0, stream>>>(dst, mk1, ev1, dn1, E, H1);
  edge_aggregate_kernel<<<blk((size_t)E * 32, 256), 256, 0, stream>>>(src, dst, z1, ev1, dn1, agg1, E, H1, F1);
  elu_bias_kernel<<<blk((size_t)N * HF1, 256), 256, 0, stream>>>(agg1, b1, (size_t)N * HF1, HF1);

  // ---- layer 2 (h = agg1 after in-place ELU) ----
  pack_b_kernel<<<blk((size_t)nt2 * ks2 * 32, 256), 256, 0, stream>>>(W2, Bp2, C, HF1);
  gemm_f32_wmma<<<N / 16, 128, 0, stream>>>(agg1, Bp2, z2, C, HF1);
  attn_coef_kernel<<<blk((size_t)N * 32, 256), 256, 0, stream>>>(z2, al2, ar2, el2, er2, N, 1, C);
  edge_score_kernel<<<blk((size_t)E, 256), 256, 0, stream>>>(src, dst, el2, er2, ev2, mk2, E, 1);
  edge_exp_kernel<<<blk((size_t)E, 256), 256, 0, stream>>>(dst, mk2, ev2, dn2, E, 1);
  edge_aggregate_kernel<<<blk((size_t)E * 32, 256), 256, 0, stream>>>(src, dst, z2, ev2, dn2, out, E, 1, C);
  bias_add_kernel<<<blk((size_t)N * C, 256), 256, 0, stream>>>(out, b2, (size_t)N * C, C);
}